// QuanvolutionFilter_65481071408030
// MI455X (gfx1250) — compile-verified
//
#include <hip/hip_runtime.h>

typedef float v2f __attribute__((ext_vector_type(2)));
typedef float v4f __attribute__((ext_vector_type(4)));
typedef float v8f __attribute__((ext_vector_type(8)));

#define N_OPS 8

// ================= setup kernel ==================
// Reproduces numpy RandomState(0) circuit structure (MT19937 + masked-rejection
// randint), builds the 16x16 complex unitary U from op_angles, and writes a
// lane-swizzled WMMA A-matrix pack (32 lanes x 16 floats: 8 x Ur, 8 x Ui) to ws.
__global__ void quanv_setup(const float* __restrict__ op_angles,
                            float* __restrict__ apack) {
  if (threadIdx.x != 0 || blockIdx.x != 0) return;

  // ---- MT19937 (numpy legacy rk_seed(0) == init_genrand(0)) ----
  unsigned mt[624];
  mt[0] = 0u;
  for (int i = 1; i < 624; ++i)
    mt[i] = 1812433253u * (mt[i - 1] ^ (mt[i - 1] >> 30)) + (unsigned)i;
  int mti = 624;
  auto draw = [&]() -> unsigned {
    if (mti >= 624) {
      for (int i = 0; i < 624; ++i) {
        unsigned y = (mt[i] & 0x80000000u) | (mt[(i + 1) % 624] & 0x7fffffffu);
        unsigned v = mt[(i + 397) % 624] ^ (y >> 1);
        if (y & 1u) v ^= 0x9908b0dfu;
        mt[i] = v;
      }
      mti = 0;
    }
    unsigned y = mt[mti++];
    y ^= y >> 11;
    y ^= (y << 7) & 0x9d2c5680u;
    y ^= (y << 15) & 0xefc60000u;
    y ^= y >> 18;
    return y;
  };
  auto rint = [&](unsigned mx) -> unsigned {  // uniform [0,mx], numpy masked rejection
    unsigned mask = mx;
    mask |= mask >> 1; mask |= mask >> 2; mask |= mask >> 4;
    mask |= mask >> 8; mask |= mask >> 16;
    unsigned v;
    do { v = draw() & mask; } while (v > mx);
    return v;
  };

  int rxw[N_OPS], cc[N_OPS], ct[N_OPS];
  for (int k = 0; k < N_OPS; ++k) rxw[k] = (int)rint(3u);
  for (int k = 0; k < N_OPS; ++k) cc[k] = (int)rint(3u);
  for (int k = 0; k < N_OPS; ++k) ct[k] = (cc[k] + 1 + (int)rint(2u)) & 3;

  // ---- build U = prod_k CNOT_k * RXfull_k  (wire 0 = bit 3, MSB) ----
  float Ur[16][16], Ui[16][16];
  for (int i = 0; i < 16; ++i)
    for (int j = 0; j < 16; ++j) {
      Ur[i][j] = (i == j) ? 1.0f : 0.0f;
      Ui[i][j] = 0.0f;
    }
  for (int k = 0; k < N_OPS; ++k) {
    // RX(theta) on wire rxw[k]: U'[i,:] = c*U[i,:] + (-i s)*U[i^m,:]
    int m = 1 << (3 - rxw[k]);
    float half = op_angles[k] * 0.5f;
    float c = cosf(half), s = sinf(half);
    for (int i = 0; i < 16; ++i) {
      if (i & m) continue;
      int i2 = i | m;
      for (int col = 0; col < 16; ++col) {
        float ar = Ur[i][col], ai = Ui[i][col];
        float br = Ur[i2][col], bi = Ui[i2][col];
        Ur[i][col]  = c * ar + s * bi;   // c*a + (-i s)*b
        Ui[i][col]  = c * ai - s * br;
        Ur[i2][col] = c * br + s * ai;   // c*b + (-i s)*a
        Ui[i2][col] = c * bi - s * ar;
      }
    }
    // CNOT(ctrl, tgt): row permutation U'[j,:] = U[j ^ (cbit(j)<<tgt), :]
    int mc = 1 << (3 - cc[k]);
    int mtk = 1 << (3 - ct[k]);
    for (int j = 0; j < 16; ++j) {
      if ((j & mc) && !(j & mtk)) {
        int j2 = j | mtk;
        for (int col = 0; col < 16; ++col) {
          float tr = Ur[j][col]; Ur[j][col] = Ur[j2][col]; Ur[j2][col] = tr;
          float ti = Ui[j][col]; Ui[j][col] = Ui[j2][col]; Ui[j2][col] = ti;
        }
      }
    }
  }

  // ---- A-matrix pack for V_WMMA_F32_16X16X4_F32 ----
  // A 16x4 layout: lanes 0-15 M=0..15 hold K={0,1}; lanes 16-31 hold K={2,3}.
  // apack[lane][0..7]  = Ur chunks j=0..3 (2 floats each)
  // apack[lane][8..15] = Ui chunks j=0..3
  for (int L = 0; L < 32; ++L) {
    int mrow = L & 15;
    int koff = (L < 16) ? 0 : 2;
    for (int j = 0; j < 4; ++j) {
      apack[L * 16 + 2 * j + 0]     = Ur[mrow][4 * j + koff + 0];
      apack[L * 16 + 2 * j + 1]     = Ur[mrow][4 * j + koff + 1];
      apack[L * 16 + 8 + 2 * j + 0] = Ui[mrow][4 * j + koff + 0];
      apack[L * 16 + 8 + 2 * j + 1] = Ui[mrow][4 * j + koff + 1];
    }
  }
}

// Half-wave swap (lane <-> lane^16) via v_permlanex16_b32, identity selectors.
// Keeps the cross-lane reduction in the VALU pipe (no DS traffic / DScnt waits).
__device__ __forceinline__ float xswap16(float v) {
  int i = __builtin_amdgcn_permlanex16(
      __float_as_int(v), __float_as_int(v), 0x76543210, 0xfedcba98, false, false);
  return __int_as_float(i);
}

// ================= main kernel ==================
// One wave handles TPW tiles of 16 quantum calls each. Per tile:
//   - each lane loads its call's 2x8 x-region (non-temporal), 4 conv dots, 4 sincos
//   - builds B (state^T, 16x16 f32) chunk-by-chunk in v2f registers
//   - 8 x V_WMMA_F32_16X16X4_F32: psi_r^T = Ur @ state^T, psi_i^T = Ui @ state^T
//   - probs = psi_r^2 + psi_i^2 (D layout: lane=call, vgpr=basis row)
//   - per-lane signed 8-way reduction + permlanex16 half-swap -> <Z_w>
#define TPW 4

__global__ __launch_bounds__(256) void quanv_main(
    const float* __restrict__ x, const float* __restrict__ conv_w,
    const float* __restrict__ apack, float* __restrict__ out, int total_tiles) {
  const int lane = threadIdx.x & 31;
  const int wave = blockIdx.x * (blockDim.x >> 5) + (threadIdx.x >> 5);
  const int nl = lane & 15;
  const bool hi = lane >= 16;

  // Load per-lane A-matrix registers (reused by every wave; keep cacheable).
  const v4f* ap = (const v4f*)(apack + lane * 16);
  v4f a0 = ap[0], a1 = ap[1], a2 = ap[2], a3 = ap[3];
  v2f Ar0 = {a0.x, a0.y}, Ar1 = {a0.z, a0.w}, Ar2 = {a1.x, a1.y}, Ar3 = {a1.z, a1.w};
  v2f Ai0 = {a2.x, a2.y}, Ai1 = {a2.z, a2.w}, Ai2 = {a3.x, a3.y}, Ai3 = {a3.z, a3.w};

  const float w00 = conv_w[0], w01 = conv_w[1], w10 = conv_w[2], w11 = conv_w[3];

  for (int t = 0; t < TPW; ++t) {
    int tile = wave * TPW + t;          // wave-uniform
    if (tile >= total_tiles) break;     // uniform branch, EXEC stays full
    int cb = tile << 4;                 // first call of tile
    int b = cb >> 12;                   // 4096 calls per image
    int rem = cb & 4095;
    int h = rem >> 5;                   // conv row (0..127)
    int gb = rem & 31;                  // group base within row (0 or 16)

    // x region for this lane's call: rows 2h..2h+1, cols 8g..8g+7 (streamed once)
    size_t xoff = ((size_t)b << 16) + ((size_t)(2 * h) << 8) + (size_t)(8 * (gb + nl));
    const float* xr = x + xoff;
    v4f r0a = __builtin_nontemporal_load((const v4f*)(xr));
    v4f r0b = __builtin_nontemporal_load((const v4f*)(xr + 4));
    v4f r1a = __builtin_nontemporal_load((const v4f*)(xr + 256));
    v4f r1b = __builtin_nontemporal_load((const v4f*)(xr + 260));

    // 2x2 stride-2 conv -> 4 encoding angles
    float v0 = w00 * r0a.x + w01 * r0a.y + w10 * r1a.x + w11 * r1a.y;
    float v1 = w00 * r0a.z + w01 * r0a.w + w10 * r1a.z + w11 * r1a.w;
    float v2 = w00 * r0b.x + w01 * r0b.y + w10 * r1b.x + w11 * r1b.y;
    float v3 = w00 * r0b.z + w01 * r0b.w + w10 * r1b.z + w11 * r1b.w;

    float s0, c0, s1, c1, s2, c2, s3, c3;
    __sincosf(0.5f * v0, &s0, &c0);
    __sincosf(0.5f * v1, &s1, &c1);
    __sincosf(0.5f * v2, &s2, &c2);
    __sincosf(0.5f * v3, &s3, &c3);

    // state[k] = f0[k3]*f1[k2]*f2[k1]*f3[k0]; B layout mirrors A:
    // lanes 0-15 hold K%4 = {0,1}, lanes 16-31 hold K%4 = {2,3}; N = lane%16.
    float f2sel = hi ? s2 : c2;                 // k bit1
    float q0 = f2sel * c3, q1 = f2sel * s3;     // k bit0 = 0 / 1
    float p0 = c0 * c1, p1 = c0 * s1, p2 = s0 * c1, p3 = s0 * s1;  // k bits [3:2]
    v2f B0 = {p0 * q0, p0 * q1};
    v2f B1 = {p1 * q0, p1 * q1};
    v2f B2 = {p2 * q0, p2 * q1};
    v2f B3 = {p3 * q0, p3 * q1};

    v8f Dr = {0.f, 0.f, 0.f, 0.f, 0.f, 0.f, 0.f, 0.f};
    v8f Di = {0.f, 0.f, 0.f, 0.f, 0.f, 0.f, 0.f, 0.f};
    // psi_r^T = Ur @ state^T ; psi_i^T = Ui @ state^T  (K=16 via 4 chunks)
    Dr = __builtin_amdgcn_wmma_f32_16x16x4_f32(false, Ar0, false, B0, (short)0, Dr, false, false);
    Di = __builtin_amdgcn_wmma_f32_16x16x4_f32(false, Ai0, false, B0, (short)0, Di, false, false);
    Dr = __builtin_amdgcn_wmma_f32_16x16x4_f32(false, Ar1, false, B1, (short)0, Dr, false, false);
    Di = __builtin_amdgcn_wmma_f32_16x16x4_f32(false, Ai1, false, B1, (short)0, Di, false, false);
    Dr = __builtin_amdgcn_wmma_f32_16x16x4_f32(false, Ar2, false, B2, (short)0, Dr, false, false);
    Di = __builtin_amdgcn_wmma_f32_16x16x4_f32(false, Ai2, false, B2, (short)0, Di, false, false);
    Dr = __builtin_amdgcn_wmma_f32_16x16x4_f32(false, Ar3, false, B3, (short)0, Dr, false, false);
    Di = __builtin_amdgcn_wmma_f32_16x16x4_f32(false, Ai3, false, B3, (short)0, Di, false, false);

    // probs for basis rows m = vgpr + (hi ? 8 : 0), column n = this lane's call
    float pb0 = Dr[0] * Dr[0] + Di[0] * Di[0];
    float pb1 = Dr[1] * Dr[1] + Di[1] * Di[1];
    float pb2 = Dr[2] * Dr[2] + Di[2] * Di[2];
    float pb3 = Dr[3] * Dr[3] + Di[3] * Di[3];
    float pb4 = Dr[4] * Dr[4] + Di[4] * Di[4];
    float pb5 = Dr[5] * Dr[5] + Di[5] * Di[5];
    float pb6 = Dr[6] * Dr[6] + Di[6] * Di[6];
    float pb7 = Dr[7] * Dr[7] + Di[7] * Di[7];

    float a01 = pb0 + pb1, a23 = pb2 + pb3, a45 = pb4 + pb5, a67 = pb6 + pb7;
    float e0 = (a01 + a23) + (a45 + a67);       // wire0 sign = bit3 of m
    if (hi) e0 = -e0;
    float e1 = (a01 + a23) - (a45 + a67);       // wire1 sign = bit2
    float e2 = (a01 - a23) + (a45 - a67);       // wire2 sign = bit1
    float e3 = (pb0 - pb1) + (pb2 - pb3) + (pb4 - pb5) + (pb6 - pb7);  // wire3 = bit0

    e0 += xswap16(e0);
    e1 += xswap16(e1);
    e2 += xswap16(e2);
    e3 += xswap16(e3);

    if (!hi) {
      v4f o = {e0, e1, e2, e3};
      __builtin_nontemporal_store(o, (v4f*)(out + (size_t)(cb + nl) * 4));
    }
  }
}

// ================= launcher ==================
extern "C" void kernel_launch(void* const* d_in, const int* in_sizes, int n_in,
                              void* d_out, int out_size, void* d_ws, size_t ws_size,
                              hipStream_t stream) {
  const float* x = (const float*)d_in[0];          // (B,1,256,256) f32
  const float* conv_w = (const float*)d_in[1];     // (1,1,2,2) f32
  const float* op_angles = (const float*)d_in[2];  // (8,) f32
  float* out = (float*)d_out;                      // (B, 16384) f32
  float* apack = (float*)d_ws;                     // 2 KB: 32 lanes x 16 floats

  int B = in_sizes[0] / 65536;                     // elements per image = 256*256
  int total_tiles = B * 256;                       // 4096 calls per image / 16

  quanv_setup<<<1, 32, 0, stream>>>(op_angles, apack);

  const int WPB = 8;                               // 256 threads = 8 wave32
  int total_waves = (total_tiles + TPW - 1) / TPW;
  int blocks = (total_waves + WPB - 1) / WPB;
  quanv_main<<<blocks, 256, 0, stream>>>(x, conv_w, apack, out, total_tiles);
}